// NanodetLoss_89343909692049
// MI455X (gfx1250) — compile-verified
//
#include <hip/hip_runtime.h>

// NanoDet GFL loss, fused single pass. Memory-bound: ~128 MB read once.
// Layouts: cls_score [B,80,H,W], bbox_pred [B,32,H,W]; n = b*4096 + h*64 + w,
// so per-channel loads across a wave are fully coalesced.

#define NCLS     80
#define RMAX     7
#define R1       8
#define HW_SHIFT 12               // H*W = 4096
#define N_TOTAL  (64 * 64 * 64)   // 262144
#define BLOCK    256
#define NBLOCKS  (N_TOTAL / BLOCK) // 1024, exact

typedef __attribute__((ext_vector_type(2))) float v2f;
typedef __attribute__((ext_vector_type(8))) float v8f;

// Full-wave (32 lane) f32 sum using V_WMMA_F32_16X16X4_F32.
// A = ones(16x4)  => D[m][n] = sum_k B[k][n]  (column sums, identical rows).
// B per lane = (p, 0): the 4 slots of column n are the x/y components of
// lanes n and n+16, so colsum(n) = p[n] + p[n+16] independent of the exact
// K<->VGPR mapping. After the WMMA every lane holds q[lane%16]; 4 xor-shuffle
// steps finish the reduction. EXEC must be all ones (guaranteed by launch).
__device__ __forceinline__ float wave_sum32(float p) {
    v2f a; a.x = 1.0f; a.y = 1.0f;   // ones A-matrix
    v2f b; b.x = p;    b.y = 0.0f;
    v8f c = {};
    c = __builtin_amdgcn_wmma_f32_16x16x4_f32(false, a, false, b,
                                              (short)0, c, false, false);
    float v = c[0];                  // q[lane % 16]
    v += __shfl_xor(v, 8);
    v += __shfl_xor(v, 4);
    v += __shfl_xor(v, 2);
    v += __shfl_xor(v, 1);
    return v;                        // wave total in every lane
}

__device__ __forceinline__ float sigmoid_from_e(float x, float e) {
    // e = exp(-|x|)
    return (x >= 0.0f) ? 1.0f / (1.0f + e) : e / (1.0f + e);
}

__global__ void __launch_bounds__(BLOCK)
nanodet_loss_main(const float* __restrict__ anchors,
                  const float* __restrict__ cls_score,
                  const float* __restrict__ bbox_pred,
                  const float* __restrict__ label_weights,
                  const float* __restrict__ bbox_targets,
                  const int*   __restrict__ labels,
                  const int*   __restrict__ stride_ptr,
                  float*       __restrict__ partials) {
    const int n = blockIdx.x * BLOCK + threadIdx.x;   // exactly N_TOTAL threads
    const float inv_s = 1.0f / (float)(*stride_ptr);  // uniform scalar load

    const int b  = n >> HW_SHIFT;
    const int hw = n & ((1 << HW_SHIFT) - 1);
    const int cls_base = ((b * NCLS)    << HW_SHIFT) + hw;
    const int box_base = ((b * 4 * R1)  << HW_SHIFT) + hw;

    const float4 anc = ((const float4*)anchors)[n];
    const float4 tg  = ((const float4*)bbox_targets)[n];
    const float cx = 0.5f * (anc.x + anc.z) * inv_s;
    const float cy = 0.5f * (anc.y + anc.w) * inv_s;
    const float tx0 = tg.x * inv_s, ty0 = tg.y * inv_s;
    const float tx1 = tg.z * inv_s, ty1 = tg.w * inv_s;

    const int  lab  = labels[n];
    const bool pos  = (lab >= 0) && (lab < NCLS);
    const float posf = pos ? 1.0f : 0.0f;

    // ---- classification pass: neg QFL sum, max logit, logit at label ----
    float max_logit = -3.402823466e38f;
    float neg_sum = 0.0f;
    float cls_at  = 0.0f;
#pragma unroll 4
    for (int c = 0; c < NCLS; ++c) {
        const float x  = cls_score[cls_base + (c << HW_SHIFT)];
        max_logit = fmaxf(max_logit, x);
        const float e   = expf(-fabsf(x));           // shared by sigmoid+softplus
        const float sig = sigmoid_from_e(x, e);
        const float sp  = fmaxf(x, 0.0f) + log1pf(e); // softplus(x)
        neg_sum += sp * sig * sig;                    // bce(x,0)*sigmoid(x)^2
        if (c == lab) cls_at = x;
    }
    const float em = expf(-fabsf(max_logit));
    const float wt = sigmoid_from_e(max_logit, em) * posf; // max sigmoid, pos only

    // ---- distribution pass: softmax integral + DFL per corner ----
    const float dist_in[4] = { cx - tx0, cy - ty0, tx1 - cx, ty1 - cy };
    float corners[4];
    float dfl_sum = 0.0f;
#pragma unroll
    for (int d = 0; d < 4; ++d) {
        float l[R1];
        float m = -3.402823466e38f;
#pragma unroll
        for (int k = 0; k < R1; ++k) {
            l[k] = bbox_pred[box_base + ((d * R1 + k) << HW_SHIFT)];
            m = fmaxf(m, l[k]);
        }
        float se = 0.0f, sk = 0.0f;
#pragma unroll
        for (int k = 0; k < R1; ++k) {
            const float e = expf(l[k] - m);
            se += e;
            sk += (float)k * e;
        }
        corners[d] = sk / se;
        const float logZ = m + logf(se);              // log-sum-exp
        float dist = fminf(fmaxf(dist_in[d], 0.0f), (float)RMAX - 0.1f);
        const int   tl = (int)floorf(dist);           // in [0,6]
        const int   tr = (tl + 1 > RMAX) ? RMAX : tl + 1;
        const float wl = (float)(tl + 1) - dist;
        const float wr = dist - (float)tl;
        dfl_sum += (logZ - l[tl]) * wl + (logZ - l[tr]) * wr; // -log_softmax
    }

    // ---- decode + GIoU ----
    const float dx0 = cx - corners[0], dy0 = cy - corners[1];
    const float dx1 = cx + corners[2], dy1 = cy + corners[3];

    const float ow = fmaxf(fminf(dx1, tx1) - fmaxf(dx0, tx0), 0.0f);
    const float oh = fmaxf(fminf(dy1, ty1) - fmaxf(dy0, ty0), 0.0f);
    const float overlap = ow * oh;
    const float ap  = (dx1 - dx0) * (dy1 - dy0);
    const float at  = (tx1 - tx0) * (ty1 - ty0);
    const float uni = fmaxf(ap + at - overlap, 1e-6f);
    const float iou = overlap / uni;
    const float ew  = fmaxf(fmaxf(dx1, tx1) - fminf(dx0, tx0), 0.0f);
    const float eh  = fmaxf(fmaxf(dy1, ty1) - fminf(dy0, ty0), 0.0f);
    const float enc = fmaxf(ew * eh, 1e-6f);
    const float giou = iou - (enc - uni) / enc;
    const float score = pos ? iou : 0.0f;

    // ---- QFL: replace label-class neg term with pos term ----
    float qfl_sum = neg_sum;
    if (pos) {
        const float ea     = expf(-fabsf(cls_at));
        const float sig_at = sigmoid_from_e(cls_at, ea);
        const float sp_at  = fmaxf(cls_at, 0.0f) + log1pf(ea);
        const float neg_at = sp_at * sig_at * sig_at;
        const float df     = score - sig_at;
        const float pos_l  = (sp_at - cls_at * score) * df * df;
        qfl_sum += pos_l - neg_at;
    }

    float p_qfl  = qfl_sum * label_weights[n];
    float p_bbox = (1.0f - giou) * wt;
    float p_dfl  = dfl_sum * wt;
    float p_wt   = wt;

    // ---- wave reduction via WMMA, then block reduction ----
    p_qfl  = wave_sum32(p_qfl);
    p_bbox = wave_sum32(p_bbox);
    p_dfl  = wave_sum32(p_dfl);
    p_wt   = wave_sum32(p_wt);

    __shared__ float4 sdata[BLOCK / 32];
    const int lane = threadIdx.x & 31;
    const int wave = threadIdx.x >> 5;
    if (lane == 0) sdata[wave] = make_float4(p_qfl, p_bbox, p_dfl, p_wt);
    __syncthreads();
    if (threadIdx.x == 0) {
        float4 acc = sdata[0];
        for (int w = 1; w < BLOCK / 32; ++w) {
            acc.x += sdata[w].x; acc.y += sdata[w].y;
            acc.z += sdata[w].z; acc.w += sdata[w].w;
        }
        ((float4*)partials)[blockIdx.x] = acc;
    }
}

// Deterministic final reduction of NBLOCKS float4 partials (single block).
__global__ void __launch_bounds__(BLOCK)
nanodet_loss_finalize(const float* __restrict__ partials,
                      const int*   __restrict__ nts_ptr,
                      float*       __restrict__ out) {
    float4 acc = make_float4(0.0f, 0.0f, 0.0f, 0.0f);
    for (int i = threadIdx.x; i < NBLOCKS; i += BLOCK) {  // uniform trip count
        const float4 p = ((const float4*)partials)[i];
        acc.x += p.x; acc.y += p.y; acc.z += p.z; acc.w += p.w;
    }
    acc.x = wave_sum32(acc.x);
    acc.y = wave_sum32(acc.y);
    acc.z = wave_sum32(acc.z);
    acc.w = wave_sum32(acc.w);

    __shared__ float4 sdata[BLOCK / 32];
    const int lane = threadIdx.x & 31;
    const int wave = threadIdx.x >> 5;
    if (lane == 0) sdata[wave] = acc;
    __syncthreads();
    if (threadIdx.x == 0) {
        float4 t = sdata[0];
        for (int w = 1; w < BLOCK / 32; ++w) {
            t.x += sdata[w].x; t.y += sdata[w].y;
            t.z += sdata[w].z; t.w += sdata[w].w;
        }
        const float inv_nts = 1.0f / (float)(*nts_ptr);
        out[0] = t.x * inv_nts;     // loss_qfl
        out[1] = 2.0f * t.y;        // loss_bbox (GIoU, weight 2.0)
        out[2] = t.z * 0.0625f;     // loss_dfl  (0.25 / 4)
        out[3] = t.w;               // sum(weight_targets)
    }
}

extern "C" void kernel_launch(void* const* d_in, const int* in_sizes, int n_in,
                              void* d_out, int out_size, void* d_ws, size_t ws_size,
                              hipStream_t stream) {
    const float* anchors       = (const float*)d_in[0];
    const float* cls_score     = (const float*)d_in[1];
    const float* bbox_pred     = (const float*)d_in[2];
    const float* label_weights = (const float*)d_in[3];
    const float* bbox_targets  = (const float*)d_in[4];
    const int*   labels        = (const int*)d_in[5];
    const int*   nts           = (const int*)d_in[6];
    const int*   stride        = (const int*)d_in[7];
    float*       partials      = (float*)d_ws;   // NBLOCKS * 4 floats = 16 KB

    nanodet_loss_main<<<NBLOCKS, BLOCK, 0, stream>>>(
        anchors, cls_score, bbox_pred, label_weights, bbox_targets,
        labels, stride, partials);
    nanodet_loss_finalize<<<1, BLOCK, 0, stream>>>(partials, nts, (float*)d_out);
}